// ReduceMask_1486058685060
// MI455X (gfx1250) — compile-verified
//
#include <hip/hip_runtime.h>
#include <hip/hip_bf16.h>
#include <stdint.h>
#include <stddef.h>

// ---------------------------------------------------------------------------
// Problem constants (match reference): N=32, H=W=1024, kernel 3x3 stride 1,
// block 16x16 -> pool window 16, pool stride 14, pad 1 -> 73x73 blocks.
// ---------------------------------------------------------------------------
#define NB      32
#define HH      1024
#define WW      1024
#define POOLK   16
#define POOLS   14
#define PAD     1
#define BH      73
#define BW      73
#define NROWS   (NB * BH)          // 2336 workgroups / block-rows
#define TOTAL   (NB * BH * BW)     // 170528 output rows
#define TPB     256
#define ROWBYTES (WW * 4)          // 4096 B: global row stride == LDS row stride

// ---------------------------------------------------------------------------
// CDNA5 async global->LDS copy (ASYNCcnt path, CDNA5 ISA).
// Per-lane: copies 16 bytes from global gaddr + IOFFSET to LDS at lds_base + IOFFSET.
// IOFFSET is applied to BOTH the LDS and global addresses (ISA 08 §4.4), so a
// constant per-row offset walks both sides in lockstep with zero address math.
// ---------------------------------------------------------------------------
template <int BYTE_OFF>
__device__ __forceinline__ void async_copy_b128_to_lds(uint32_t lds_off,
                                                       const void* gaddr) {
#if defined(__AMDGCN__)
  asm volatile("global_load_async_to_lds_b128 %0, %1, off offset:%c2"
               :: "v"(lds_off), "v"(gaddr), "i"(BYTE_OFF)
               : "memory");
#endif
}

__device__ __forceinline__ void wait_asynccnt0() {
#if defined(__AMDGCN__)
  asm volatile("s_wait_asynccnt 0x0" ::: "memory");
#endif
}

// Straight-line issue of rows [1, POOLK): offset j*4096 on both LDS & global.
template <int J>
__device__ __forceinline__ void load_rows_from1(uint32_t lds0, const void* g0) {
  if constexpr (J < POOLK) {
    async_copy_b128_to_lds<J * ROWBYTES>(lds0, g0);
    load_rows_from1<J + 1>(lds0, g0);
  }
}

// ---------------------------------------------------------------------------
// Kernel 1: per (n, by) block-row, compute 73 pooled-max "active" flags.
//   - async-load the 16 window rows into a 64 KB LDS tile (one base address,
//     15 straight-line issues with immediate offsets; row 0 predicated since
//     only by==0 can touch the top padding row)
//   - column-max over 16 rows (thread t owns cols t, t+256, t+512, t+768;
//     lane-consecutive words -> conflict-free LDS banks)
//   - 73 horizontal window maxes of 16 columns each
// Zero is a safe max identity: padding contributes 0 and the decision is
// (max > 0.5), so max(vals,0) > 0.5  <=>  max(vals) > 0.5.
// ---------------------------------------------------------------------------
__global__ void __launch_bounds__(TPB)
pool_rows_kernel(const float* __restrict__ mask,
                 int* __restrict__ flags,
                 int* __restrict__ rowCounts) {
  __shared__ float tile[POOLK][WW];   // 64 KB
  __shared__ float colmax[WW];        // 4 KB
  __shared__ int   acts[BW];

  const int wg = blockIdx.x;          // wg = n*BH + by
  const int n  = wg / BH;
  const int by = wg - n * BH;
  const int t  = threadIdx.x;
  const int r0 = by * POOLS - PAD;    // first window row (-1 only when by==0)

  // One per-lane base for global and LDS; rows advance via IOFFSET.
  const char* img = (const char*)mask + (size_t)n * HH * (size_t)ROWBYTES;
  const char* g0  = img + (ptrdiff_t)r0 * ROWBYTES + t * 16;
  const uint32_t lds0 = (uint32_t)(uintptr_t)(&tile[0][t * 4]);

  if (r0 >= 0) {                       // wave-uniform: false only for by==0
    async_copy_b128_to_lds<0>(lds0, g0);
  }
  load_rows_from1<1>(lds0, g0);        // rows 1..15 always in-bounds
  wait_asynccnt0();
  __syncthreads();

  // Column max over the 16 rows (invalid top row masked, never read as data).
  const bool row0_valid = (r0 >= 0);
#pragma unroll
  for (int k = 0; k < WW / TPB; ++k) {
    const int c = t + k * TPB;
    float m = row0_valid ? tile[0][c] : 0.0f;
#pragma unroll
    for (int j = 1; j < POOLK; ++j) {
      m = fmaxf(m, tile[j][c]);
    }
    colmax[c] = m;
  }
  __syncthreads();

  // 73 horizontal window maxes; flag layout is flat lexicographic:
  // f = (n*BH + by)*BW + bx  -- exactly the required output order.
  if (t < BW) {
    const int c0 = t * POOLS - PAD;
    float m = 0.0f;
#pragma unroll
    for (int k = 0; k < POOLK; ++k) {
      const int c = c0 + k;
      const float v = (c >= 0 && c < WW) ? colmax[c] : 0.0f;
      m = fmaxf(m, v);
    }
    const int a = (m > 0.5f) ? 1 : 0;
    flags[(size_t)wg * BW + t] = a;
    acts[t] = a;
  }
  __syncthreads();

  if (t == 0) {
    int s = 0;
#pragma unroll 1
    for (int i = 0; i < BW; ++i) s += acts[i];
    rowCounts[wg] = s;
  }
}

// ---------------------------------------------------------------------------
// Kernel 2: single-workgroup exclusive scan of the 2336 row counts.
// Also writes the total active count into the final output slot.
// ---------------------------------------------------------------------------
__global__ void __launch_bounds__(TPB)
scan_rows_kernel(const int* __restrict__ rowCounts,
                 int* __restrict__ rowOffsets,
                 int* __restrict__ totalOut,
                 int* __restrict__ outCountSlot) {
  __shared__ int partial[TPB];
  const int t = threadIdx.x;
  const int per = (NROWS + TPB - 1) / TPB;   // 10
  const int base = t * per;

  int s = 0;
  for (int i = 0; i < per; ++i) {
    const int idx = base + i;
    if (idx < NROWS) s += rowCounts[idx];
  }
  partial[t] = s;
  __syncthreads();

  if (t == 0) {
    int acc = 0;
#pragma unroll 1
    for (int i = 0; i < TPB; ++i) {
      const int v = partial[i];
      partial[i] = acc;
      acc += v;
    }
    *totalOut = acc;
    *outCountSlot = acc;   // reference's `count` (int32), last output element
  }
  __syncthreads();

  int acc = partial[t];
  for (int i = 0; i < per; ++i) {
    const int idx = base + i;
    if (idx < NROWS) {
      rowOffsets[idx] = acc;
      acc += rowCounts[idx];
    }
  }
}

// ---------------------------------------------------------------------------
// Kernel 3: order-preserving scatter of active (n, by, bx) triples.
// ---------------------------------------------------------------------------
__global__ void __launch_bounds__(128)
scatter_kernel(const int* __restrict__ flags,
               const int* __restrict__ rowOffsets,
               int* __restrict__ out) {
  __shared__ int f[BW];
  __shared__ int pfx[BW];

  const int wg = blockIdx.x;
  const int n  = wg / BH;
  const int by = wg - n * BH;
  const int t  = threadIdx.x;

  if (t < BW) f[t] = flags[(size_t)wg * BW + t];
  __syncthreads();

  if (t == 0) {
    int acc = 0;
#pragma unroll 1
    for (int i = 0; i < BW; ++i) { pfx[i] = acc; acc += f[i]; }
  }
  __syncthreads();

  if (t < BW && f[t]) {
    const int pos = rowOffsets[wg] + pfx[t];
    out[3 * pos + 0] = n;
    out[3 * pos + 1] = by;
    out[3 * pos + 2] = t;
  }
}

// ---------------------------------------------------------------------------
// Kernel 4: fill inactive tail slots with the reference's fill value (N,BH,BW).
// ---------------------------------------------------------------------------
__global__ void __launch_bounds__(TPB)
fill_tail_kernel(const int* __restrict__ total,
                 int* __restrict__ out) {
  const int count = *total;
  const int idx = blockIdx.x * blockDim.x + threadIdx.x;
  if (idx < TOTAL && idx >= count) {
    out[3 * idx + 0] = NB;
    out[3 * idx + 1] = BH;
    out[3 * idx + 2] = BW;
  }
}

// ---------------------------------------------------------------------------
// Launch: mask -> flags/rowCounts -> scan -> scatter + tail fill.
// d_out layout (int32 viewed): [TOTAL*3] sorted indices, then [1] count.
// ---------------------------------------------------------------------------
extern "C" void kernel_launch(void* const* d_in, const int* in_sizes, int n_in,
                              void* d_out, int out_size, void* d_ws, size_t ws_size,
                              hipStream_t stream) {
  (void)in_sizes; (void)n_in; (void)out_size; (void)ws_size;

  const float* mask = (const float*)d_in[0];
  int* out = (int*)d_out;                 // int32 output, bit-exact layout

  int* flags      = (int*)d_ws;           // TOTAL ints
  int* rowCounts  = flags + TOTAL;        // NROWS ints
  int* rowOffsets = rowCounts + NROWS;    // NROWS ints
  int* total      = rowOffsets + NROWS;   // 1 int

  pool_rows_kernel<<<NROWS, TPB, 0, stream>>>(mask, flags, rowCounts);
  scan_rows_kernel<<<1, TPB, 0, stream>>>(rowCounts, rowOffsets, total,
                                          out + 3 * TOTAL);
  scatter_kernel<<<NROWS, 128, 0, stream>>>(flags, rowOffsets, out);
  fill_tail_kernel<<<(TOTAL + TPB - 1) / TPB, TPB, 0, stream>>>(total, out);
}